// PyrmaidCostVolume_65541200937417
// MI455X (gfx1250) — compile-verified
//
#include <hip/hip_runtime.h>
#include <stdint.h>

#define TPB 256
#define SM_STRIDE 68   // floats per thread window: 64 data + 4 pad; 272B (16B aligned)

typedef int v4i __attribute__((ext_vector_type(4)));
typedef __attribute__((address_space(1))) v4i gas_v4i;   // global (device) AS
typedef __attribute__((address_space(3))) v4i las_v4i;   // LDS AS

// ---- CDNA5 async global->LDS helpers (builtin if present, else inline asm) ----
__device__ __forceinline__ void async_ld_b128(const float* g, float* l) {
#if defined(__has_builtin) && __has_builtin(__builtin_amdgcn_global_load_async_to_lds_b128)
  __builtin_amdgcn_global_load_async_to_lds_b128(
      (gas_v4i*)(float*)g, (las_v4i*)l, 0, 0);
#else
  uint32_t lds = (uint32_t)(uintptr_t)(__attribute__((address_space(3))) void*)l;
  asm volatile("global_load_async_to_lds_b128 %0, %1, off"
               :: "v"(lds), "v"(g) : "memory");
#endif
}

__device__ __forceinline__ void wait_async_zero() {
#if defined(__has_builtin) && __has_builtin(__builtin_amdgcn_s_wait_asynccnt)
  __builtin_amdgcn_s_wait_asynccnt(0);
#else
  asm volatile("s_wait_asynccnt 0x0" ::: "memory");
#endif
  asm volatile("" ::: "memory");  // compiler ordering fence before LDS reads
}

// Pooled pyramid tap computed on the fly from the raw level-0 window in LDS.
__device__ __forceinline__ float tapv(const float* wv, int lvl, int j, int start) {
  if (lvl == 0) {
    return wv[j - start];
  } else if (lvl == 1) {
    float2 v = *reinterpret_cast<const float2*>(wv + (2 * j - start)); // 8B aligned
    return 0.5f * (v.x + v.y);
  } else {
    float4 v = *reinterpret_cast<const float4*>(wv + (4 * j - start)); // 16B aligned
    return 0.5f * (0.5f * (v.x + v.y) + 0.5f * (v.z + v.w));
  }
}

__global__ __launch_bounds__(TPB)
void PyrmaidCostVolume_kernel(const float* __restrict__ corr,
                              const float* __restrict__ disp,
                              float* __restrict__ outv,
                              float* __restrict__ outd) {
  __shared__ float smem[TPB * SM_STRIDE];  // ~68KB

  const int tid = threadIdx.x;
  const int p   = blockIdx.x * TPB + tid;          // pixel id == ((b*96+h)*320+w)
  const int w   = p % 320;
  const int bh  = p / 320;                          // b*96+h
  const int b   = bh / 96;
  const int rem = p - b * 30720;                    // h*320 + w

  const float d   = disp[p];                        // cur_disp flat index == p
  const float co0 = (float)w - d;

  // 64-float window covering all taps of all 3 levels (proof in analysis)
  int fs = ((int)floorf(co0) - 20) & ~3;
  const int start = min(max(fs, 0), 256);

  // Stage my row window: 16 x per-lane async B128 (global 16B aligned: p*320+start, start%4==0)
  const float* rowg = corr + (size_t)p * 320 + start;
  float* myw = &smem[tid * SM_STRIDE];
#pragma unroll
  for (int k = 0; k < 16; ++k)
    async_ld_b128(rowg + 4 * k, myw + 4 * k);
  wait_async_zero();

  const float INTERVAL = (float)(4.0 / 7.0);
  const size_t obase = (size_t)b * 1382400 + (size_t)rem;  // b*45*30720 + h*320 + w

#pragma unroll
  for (int lvl = 0; lvl < 3; ++lvl) {
    const float inv      = (lvl == 0) ? 1.0f : ((lvl == 1) ? 0.5f : 0.25f);
    const float Wi1      = (float)((320 >> lvl) - 1);
    const float standard = (float)w * inv;            // arange(W)/scale (exact)
    const float dl       = d * inv;                   // cur_disp / 2^i (exact)
    const float c_old    = standard - dl;
    const float center   = fminf(fmaxf(c_old, 0.0f), Wi1);

#pragma unroll
    for (int k = 0; k < 15; ++k) {
      float cand;
      bool valid;
      if (k <= 7) {                                   // left branch, sample s=k
        float raw = (center - 4.0f) + (float)k * INTERVAL;
        valid = raw > 0.0f;                           // pre-clip validity (ref)
        cand  = fmaxf(raw, 0.0f);
      } else {                                        // right branch, s = 14-k
        cand = (center + 4.0f) - (float)(14 - k) * INTERVAL;
        float dn = standard - cand;
        valid = (dn > 0.0f) && (dn < Wi1);
      }

      float val = 0.0f;
      if (valid) {
        float cf = fminf(fmaxf(ceilf(cand),  0.0f), Wi1);
        float ff = fminf(fmaxf(floorf(cand), 0.0f), Wi1);
        float fr = cf - cand;                         // floor_rate (ref math)
        float cr = 1.0f - fr;                         // ceil_rate
        int   ci = (int)cf;
        int   fi = (int)ff;
        val = tapv(myw, lvl, ci, start) * cr + tapv(myw, lvl, fi, start) * fr;
      }

      const size_t oidx = obase + (size_t)(lvl * 15 + k) * 30720;
      outv[oidx] = val;
      outd[oidx] = (float)(k - 7) * INTERVAL + dl;    // disp_cands
    }
  }
}

extern "C" void kernel_launch(void* const* d_in, const int* in_sizes, int n_in,
                              void* d_out, int out_size, void* d_ws, size_t ws_size,
                              hipStream_t stream) {
  const float* corr = (const float*)d_in[0];  // [8,96,320,320]
  const float* disp = (const float*)d_in[1];  // [8,1,96,320]
  float* outv = (float*)d_out;                          // [8,45,96,320]
  float* outd = outv + (size_t)8 * 45 * 96 * 320;       // [8,45,96,320]

  const int total = 8 * 96 * 320;                       // 245760 pixels
  dim3 grid(total / TPB), block(TPB);                   // 960 blocks, no tail
  hipLaunchKernelGGL(PyrmaidCostVolume_kernel, grid, block, 0, stream,
                     corr, disp, outv, outd);
}